// GAT_23063974380115
// MI455X (gfx1250) — compile-verified
//
#include <hip/hip_runtime.h>
#include <math.h>

#define NEG_SLOPE 0.2f

typedef __attribute__((ext_vector_type(2))) float v2f;
typedef __attribute__((ext_vector_type(8))) float v8f;

__device__ __forceinline__ float lrelu(float x) {
    return x > 0.0f ? x : NEG_SLOPE * x;
}

// CAS-based float atomic max (handles negatives; values here are finite)
__device__ __forceinline__ void atomicMaxF(float* addr, float val) {
    int* ai = (int*)addr;
    int old = __float_as_int(*addr);
    while (__int_as_float(old) < val) {
        int assumed = old;
        old = atomicCAS(ai, assumed, __float_as_int(val));
        if (old == assumed) break;
    }
}

// ---------------------------------------------------------------------------
// FP32 WMMA GEMM: C[M,NC] = A[M,K] @ B[K,NC], one wave per 16x16 output tile.
// Uses V_WMMA_F32_16X16X4_F32 (exact fp32, K-step 4, 32 steps for K=128).
// A 16x4 layout: lanes 0-15 M=0..15 {K=k0,k0+1}, lanes 16-31 {K=k0+2,k0+3}.
// B 4x16 layout: lane%16 = N, VGPR0={K=k0|k0+2}, VGPR1={K=k0+1|k0+3}.
// C 16x16 layout: VGPR i -> M = i + 8*(lane>=16), N = lane%16.
// ---------------------------------------------------------------------------
__global__ void gat_wmma_gemm(const float* __restrict__ A,
                              const float* __restrict__ B,
                              float* __restrict__ C,
                              int M, int K, int NC) {
    const int wave = threadIdx.x >> 5;
    const int lane = threadIdx.x & 31;
    const int ntn  = NC >> 4;
    const int tile = blockIdx.x * (blockDim.x >> 5) + wave;
    const int total = (M >> 4) * ntn;
    if (tile >= total) return;              // wave-uniform: EXEC stays all-1s
    const int mt = tile / ntn;
    const int nt = tile - mt * ntn;
    const int half = lane >> 4;             // 0 or 1
    const int l    = lane & 15;

    const float* arow = A + (size_t)(mt * 16 + l) * K;       // lda == K
    const float* bcol = B + nt * 16 + l;                     // ldb == NC

    v8f acc = {};
#pragma unroll 4
    for (int k0 = 0; k0 < K; k0 += 4) {
        const int ka = k0 + 2 * half;
        v2f a, b;
        a.x = arow[ka];
        a.y = arow[ka + 1];
        b.x = bcol[(size_t)ka * NC];
        b.y = bcol[(size_t)(ka + 1) * NC];
        acc = __builtin_amdgcn_wmma_f32_16x16x4_f32(
            false, a, false, b, (short)0, acc, false, false);
    }

    float* crow = C + (size_t)(mt * 16 + half * 8) * NC + nt * 16 + l;
#pragma unroll
    for (int i = 0; i < 8; ++i)
        crow[(size_t)i * NC] = acc[i];
}

// el[n,h] = sum_d feat[n,h,d]*al[h,d]; er analogous
template <int H, int D>
__global__ void gat_scores(const float* __restrict__ feat,
                           const float* __restrict__ al,
                           const float* __restrict__ ar,
                           float* __restrict__ el, float* __restrict__ er,
                           int N) {
    int t = blockIdx.x * blockDim.x + threadIdx.x;
    if (t >= N * H) return;
    int n = t / H, h = t - n * H;
    const float* f = feat + (size_t)n * (H * D) + h * D;
    const float* a = al + h * D;
    const float* r = ar + h * D;
    float sl = 0.0f, sr = 0.0f;
#pragma unroll 8
    for (int d = 0; d < D; ++d) { sl += f[d] * a[d]; sr += f[d] * r[d]; }
    el[t] = sl;
    er[t] = sr;
}

// rst[n,:] = bias (so segment-sum lands on rst+b); m=-inf; den=0
__global__ void gat_init(float* __restrict__ rst, const float* __restrict__ b,
                         float* __restrict__ m, float* __restrict__ den,
                         int total, int HD, int NH) {
    int t = blockIdx.x * blockDim.x + threadIdx.x;
    if (t < total) rst[t] = b[t % HD];
    if (t < NH) { m[t] = -INFINITY; den[t] = 0.0f; }
}

template <int H>
__global__ void gat_edge_max(const float* __restrict__ el,
                             const float* __restrict__ er,
                             const int* __restrict__ src,
                             const int* __restrict__ dst,
                             float* __restrict__ m, int E) {
    int t = blockIdx.x * blockDim.x + threadIdx.x;
    if (t >= E * H) return;
    int e = t / H, h = t - e * H;
    int s = src[e], d = dst[e];
    float v = lrelu(el[s * H + h] + er[d * H + h]);
    atomicMaxF(&m[d * H + h], v);
}

template <int H>
__global__ void gat_edge_sum(const float* __restrict__ el,
                             const float* __restrict__ er,
                             const float* __restrict__ m,
                             const int* __restrict__ src,
                             const int* __restrict__ dst,
                             float* __restrict__ den, int E) {
    int t = blockIdx.x * blockDim.x + threadIdx.x;
    if (t >= E * H) return;
    int e = t / H, h = t - e * H;
    int s = src[e], d = dst[e];
    float v = lrelu(el[s * H + h] + er[d * H + h]);
    atomicAdd(&den[d * H + h], __expf(v - m[d * H + h]));
}

// One wave per edge; lane owns H*D/32 contiguous features (coalesced b128
// gather of feat[src], fp32 atomic adds into rst[dst], all L2-resident).
template <int H, int D>
__global__ void gat_edge_aggr(const float* __restrict__ feat,
                              const float* __restrict__ el,
                              const float* __restrict__ er,
                              const float* __restrict__ m,
                              const float* __restrict__ den,
                              const int* __restrict__ src,
                              const int* __restrict__ dst,
                              float* __restrict__ rst, int E) {
    constexpr int HD  = H * D;
    constexpr int EPL = HD / 32;            // 4 (layer1) or 2 (layer2)
    static_assert(D % EPL == 0, "lane spans a single head");
    const int wave = threadIdx.x >> 5;
    const int lane = threadIdx.x & 31;
    const int e = blockIdx.x * (blockDim.x >> 5) + wave;
    if (e >= E) return;
    const int base = lane * EPL;
    const int h = base / D;
    const int s = src[e], d = dst[e];
    const float v = lrelu(el[s * H + h] + er[d * H + h]);
    const float alpha = __expf(v - m[d * H + h]) / den[d * H + h];
    const float* fs = feat + (size_t)s * HD + base;
    float* rd = rst + (size_t)d * HD + base;
#pragma unroll
    for (int j = 0; j < EPL; ++j)
        atomicAdd(&rd[j], fs[j] * alpha);
}

__global__ void gat_relu(float* __restrict__ x, int n) {
    int t = blockIdx.x * blockDim.x + threadIdx.x;
    if (t < n) x[t] = fmaxf(x[t], 0.0f);
}

extern "C" void kernel_launch(void* const* d_in, const int* in_sizes, int n_in,
                              void* d_out, int out_size, void* d_ws, size_t ws_size,
                              hipStream_t stream) {
    const float* x   = (const float*)d_in[0];
    const int*   src = (const int*)d_in[1];
    const int*   dst = (const int*)d_in[2];
    const float* W1  = (const float*)d_in[3];
    const float* al1 = (const float*)d_in[4];
    const float* ar1 = (const float*)d_in[5];
    const float* b1  = (const float*)d_in[6];
    const float* W2  = (const float*)d_in[7];
    const float* al2 = (const float*)d_in[8];
    const float* ar2 = (const float*)d_in[9];
    const float* b2  = (const float*)d_in[10];
    float* out = (float*)d_out;

    constexpr int H1 = 4, D1 = 32, HD1 = 128;      // layer 1 heads
    constexpr int H2 = 1, D2 = 64, HD2 = 64;       // layer 2
    const int FIN = in_sizes[3] / HD1;             // 128
    const int N   = in_sizes[0] / FIN;             // 100000
    const int E   = in_sizes[1];                   // 1.6M

    // Workspace layout (floats)
    float* ws    = (float*)d_ws;
    float* feat1 = ws;                             // N*128, later reused as feat2 (N*64)
    float* hbuf  = feat1 + (size_t)N * HD1;        // N*128: rst1(+b1) -> relu -> layer2 input
    float* el1   = hbuf + (size_t)N * HD1;         // N*H1
    float* er1   = el1 + (size_t)N * H1;
    float* m1    = er1 + (size_t)N * H1;
    float* den1  = m1  + (size_t)N * H1;
    float* el2   = den1 + (size_t)N * H1;          // N
    float* er2   = el2 + N;
    float* m2    = er2 + N;
    float* den2  = m2  + N;
    float* feat2 = feat1;                          // alias: feat1 dead after layer-1 aggr

    const int BLK = 256;
    const int WPB = BLK / 32;                      // waves per block

    // ---------------- Layer 1 ----------------
    {
        int tiles = (N / 16) * (HD1 / 16);
        gat_wmma_gemm<<<(tiles + WPB - 1) / WPB, BLK, 0, stream>>>(x, W1, feat1, N, FIN, HD1);
    }
    gat_scores<H1, D1><<<(N * H1 + BLK - 1) / BLK, BLK, 0, stream>>>(feat1, al1, ar1, el1, er1, N);
    gat_init<<<((size_t)N * HD1 + BLK - 1) / BLK, BLK, 0, stream>>>(hbuf, b1, m1, den1, N * HD1, HD1, N * H1);
    gat_edge_max<H1><<<((size_t)E * H1 + BLK - 1) / BLK, BLK, 0, stream>>>(el1, er1, src, dst, m1, E);
    gat_edge_sum<H1><<<((size_t)E * H1 + BLK - 1) / BLK, BLK, 0, stream>>>(el1, er1, m1, src, dst, den1, E);
    gat_edge_aggr<H1, D1><<<(E + WPB - 1) / WPB, BLK, 0, stream>>>(feat1, el1, er1, m1, den1, src, dst, hbuf, E);
    gat_relu<<<((size_t)N * HD1 + BLK - 1) / BLK, BLK, 0, stream>>>(hbuf, N * HD1);

    // ---------------- Layer 2 ----------------
    {
        int tiles = (N / 16) * (HD2 / 16);
        gat_wmma_gemm<<<(tiles + WPB - 1) / WPB, BLK, 0, stream>>>(hbuf, W2, feat2, N, HD1, HD2);
    }
    gat_scores<H2, D2><<<(N * H2 + BLK - 1) / BLK, BLK, 0, stream>>>(feat2, al2, ar2, el2, er2, N);
    gat_init<<<((size_t)N * HD2 + BLK - 1) / BLK, BLK, 0, stream>>>(out, b2, m2, den2, N * HD2, HD2, N * H2);
    gat_edge_max<H2><<<((size_t)E * H2 + BLK - 1) / BLK, BLK, 0, stream>>>(el2, er2, src, dst, m2, E);
    gat_edge_sum<H2><<<((size_t)E * H2 + BLK - 1) / BLK, BLK, 0, stream>>>(el2, er2, m2, src, dst, den2, E);
    gat_edge_aggr<H2, D2><<<(E + WPB - 1) / WPB, BLK, 0, stream>>>(feat2, el2, er2, m2, den2, src, dst, out, E);
}